// AdaptiveConvND_23304492548278
// MI455X (gfx1250) — compile-verified
//
#include <hip/hip_runtime.h>
#include <hip/hip_bf16.h>
#include <math.h>

#define HH 8
#define PP 16
#define SS 33
#define BB 2
#define LL 4096
#define CC 512
#define MM (BB*LL)   // 8192 rows

typedef float v2f __attribute__((ext_vector_type(2)));
typedef float v8f __attribute__((ext_vector_type(8)));

__device__ __forceinline__ float sigm_(float z){ return 1.0f/(1.0f+expf(-z)); }
__device__ __forceinline__ float silu_(float z){ return z*sigm_(z); }

// Out[M x Nout] = act( (A .* Amul) @ W^T + bias ), A: MxK row-major, W: NxK row-major.
// Block = 256 threads = 8 waves, wave grid 4(M) x 2(N), each wave a 32x32 tile
// (2x2 V_WMMA_F32_16X16X4_F32 register blocking). Block tile: 128(M) x 64(N).
// TAIL=true: N-tail handled by pointer clamp + 0/1 float mask (no exec divergence).
// TAIL=false: N is a multiple of 64 -> raw loads, zero masking overhead.
// ACT: 0=none, 1=silu, 2=sigmoid
template<int ACT, bool HASMUL, bool TAIL>
__global__ void gemm_wmma_kernel(const float* __restrict__ A,
                                 const float* __restrict__ Amul,
                                 const float* __restrict__ W,
                                 const float* __restrict__ bias,
                                 float* __restrict__ Out,
                                 int M, int N, int K, int Nout)
{
  const int lane = threadIdx.x & 31;
  const int wv   = threadIdx.x >> 5;   // 0..7
  const int wm   = wv & 3;             // wave M index
  const int wn   = wv >> 2;            // wave N index
  const int grp  = lane >> 4;          // half-wave
  const int lr   = lane & 15;
  const int m0 = blockIdx.x * 128 + wm * 32;
  const int n0 = blockIdx.y * 64  + wn * 32;

  const int ma0 = m0 + lr,      ma1 = m0 + 16 + lr;
  const int nb0 = n0 + lr,      nb1 = n0 + 16 + lr;
  const float* arow0 = A + (size_t)ma0 * K;
  const float* arow1 = A + (size_t)ma1 * K;
  const float* mul0  = HASMUL ? (Amul + (size_t)ma0 * K) : nullptr;
  const float* mul1  = HASMUL ? (Amul + (size_t)ma1 * K) : nullptr;
  const float bm0 = (!TAIL || nb0 < N) ? 1.0f : 0.0f;
  const float bm1 = (!TAIL || nb1 < N) ? 1.0f : 0.0f;
  const float* wrow0 = W + (size_t)(TAIL ? (nb0 < N ? nb0 : N-1) : nb0) * K;
  const float* wrow1 = W + (size_t)(TAIL ? (nb1 < N ? nb1 : N-1) : nb1) * K;

  v8f acc00 = {0.f,0.f,0.f,0.f,0.f,0.f,0.f,0.f};
  v8f acc01 = acc00, acc10 = acc00, acc11 = acc00;

  #pragma unroll 2
  for (int k0 = 0; k0 < K; k0 += 4) {
    const int ka = k0 + grp*2;          // ISA 16x4 A layout: VGPR v <-> K = grp*2+v
    v2f a0, a1, b0, b1;
    a0[0] = arow0[ka]; a0[1] = arow0[ka+1];
    a1[0] = arow1[ka]; a1[1] = arow1[ka+1];
    if (HASMUL) {
      a0[0] *= mul0[ka]; a0[1] *= mul0[ka+1];
      a1[0] *= mul1[ka]; a1[1] *= mul1[ka+1];
    }
    b0[0] = wrow0[ka]; b0[1] = wrow0[ka+1];
    b1[0] = wrow1[ka]; b1[1] = wrow1[ka+1];
    if (TAIL) {
      b0[0] *= bm0; b0[1] *= bm0;
      b1[0] *= bm1; b1[1] *= bm1;
    }
    acc00 = __builtin_amdgcn_wmma_f32_16x16x4_f32(false, a0, false, b0, (short)0, acc00, false, false);
    acc01 = __builtin_amdgcn_wmma_f32_16x16x4_f32(false, a0, false, b1, (short)0, acc01, false, false);
    acc10 = __builtin_amdgcn_wmma_f32_16x16x4_f32(false, a1, false, b0, (short)0, acc10, false, false);
    acc11 = __builtin_amdgcn_wmma_f32_16x16x4_f32(false, a1, false, b1, (short)0, acc11, false, false);
  }

  // Epilogue: C layout per frag: VGPR r -> M = grp*8 + r, N = lr
  v8f accs[2][2] = {{acc00, acc01}, {acc10, acc11}};
  #pragma unroll
  for (int ni = 0; ni < 2; ++ni) {
    const int ncol = n0 + ni*16 + lr;
    if (ncol < Nout) {
      const float bn = bias ? bias[ncol] : 0.0f;
      #pragma unroll
      for (int mi = 0; mi < 2; ++mi) {
        #pragma unroll
        for (int r = 0; r < 8; ++r) {
          const int m = m0 + mi*16 + grp*8 + r;
          float v = accs[mi][ni][r] + bn;
          if (ACT == 1)      v = silu_(v);
          else if (ACT == 2) v = sigm_(v);
          Out[(size_t)m * Nout + ncol] = v;
        }
      }
    }
  }
}

// Per (b,l): freq/phase/decay from wave, qk[h] = sum_p q*Wk, sample indices,
// factorized logits rel*qk*SCALE, masked softmax * envelope, renormalize,
// entropy partial per row. Block = 128 threads.
__global__ void attn_weights_kernel(const float* __restrict__ wave, // M x 24 (silu'd)
                                    const float* __restrict__ q,    // M x 128 (silu'd)
                                    const float* __restrict__ Wk,   // 16
                                    int*   __restrict__ idxs,       // M x 33
                                    float* __restrict__ aw,         // M x 8 x 33
                                    float* __restrict__ ent)        // M
{
  const int bl = blockIdx.x;
  const int l  = bl % LL;
  const int t  = threadIdx.x;
  __shared__ float s_q[128];
  __shared__ float s_qk[HH], s_freq[HH], s_phase[HH], s_decay[HH];
  __shared__ float s_fa, s_pa;
  __shared__ float s_valid[SS];
  __shared__ float s_ent[HH];

  s_q[t] = q[(size_t)bl*128 + t] * Wk[t & 15];
  __syncthreads();
  if (t < HH) {
    float a = 0.f;
    #pragma unroll
    for (int p = 0; p < 16; ++p) a += s_q[t*16 + p];
    s_qk[t] = a;
    const float* wvp = wave + (size_t)bl*24;
    s_freq[t]  = sigm_(wvp[t])      * 15.0f + 1.0f;
    s_phase[t] = tanhf(wvp[8 + t])  * 16.0f;
    s_decay[t] = sigm_(wvp[16 + t]) * 9.5f  + 0.5f;
  }
  __syncthreads();
  if (t == 0) {
    float fa = 0.f, pa = 0.f;
    #pragma unroll
    for (int h = 0; h < HH; ++h) { fa += s_freq[h]; pa += s_phase[h]; }
    s_fa = fa * 0.125f; s_pa = pa * 0.125f;
  }
  __syncthreads();
  if (t < SS) {
    const float off = (float)(t - 16);
    const float pos = (float)l + off * s_fa + s_pa;
    const bool vld = (pos >= 0.0f) && (pos < (float)LL);
    int id = (int)pos;                       // trunc-toward-zero like astype(int32)
    id = id < 0 ? 0 : (id > LL-1 ? LL-1 : id);
    idxs[(size_t)bl*SS + t] = id;
    s_valid[t] = vld ? 1.0f : 0.0f;
  }
  __syncthreads();
  if (t < HH) {
    const int h = t;
    const float qkh = s_qk[h] * 0.25f;       // SCALE = P^-0.5 = 0.25
    const float fr  = s_freq[h];
    const float dc  = fmaxf(s_decay[h], 0.1f);
    float mx = -__builtin_inff();
    for (int s = 0; s < SS; ++s)
      if (s_valid[s] > 0.5f)
        mx = fmaxf(mx, fabsf((float)(s-16)) * fr * qkh);

    float* awrow = aw + ((size_t)bl*HH + h) * SS;
    float av[SS];
    float eacc = 0.f;
    if (!(mx > -__builtin_inff())) {         // no valid sample
      for (int s = 0; s < SS; ++s) awrow[s] = 0.0f;
      s_ent[h] = 0.0f;
    } else {
      float sexp = 0.f;
      for (int s = 0; s < SS; ++s)
        if (s_valid[s] > 0.5f)
          sexp += expf(fabsf((float)(s-16)) * fr * qkh - mx);
      const float inv = 1.0f / sexp;
      float asum = 0.f;
      for (int s = 0; s < SS; ++s) {
        float a = 0.f;
        if (s_valid[s] > 0.5f) {
          const float rel = fabsf((float)(s-16)) * fr;
          a = expf(rel * qkh - mx) * inv * expf(-rel / dc);
        }
        av[s] = a;
        asum += a;
      }
      const float rn = 1.0f / (asum + 1e-8f);
      for (int s = 0; s < SS; ++s) {
        const float a = av[s] * rn;
        awrow[s] = a;
        eacc += a * logf(a + 1e-8f);         // = -(entropy contribution)
      }
      s_ent[h] = eacc;
    }
  }
  __syncthreads();
  if (t == 0) {
    float e = 0.f;
    #pragma unroll
    for (int h = 0; h < HH; ++h) e += s_ent[h];
    ent[bl] = e;
  }
}

// out_attn[bl, c] = sum_s aw[bl, h(c), s] * x[b, idx[bl,s], c]
__global__ void gather_kernel(const float* __restrict__ x,
                              const float* __restrict__ aw,
                              const int*   __restrict__ idxs,
                              float* __restrict__ out_attn)
{
  const int bl = blockIdx.x;
  const int b  = bl / LL;
  const int t  = threadIdx.x;                // 256
  __shared__ float s_aw[HH*SS];
  __shared__ int   s_idx[SS];
  for (int j = t; j < HH*SS; j += 256) s_aw[j] = aw[(size_t)bl*HH*SS + j];
  if (t < SS) s_idx[t] = idxs[(size_t)bl*SS + t];
  __syncthreads();
  const float* xb = x + (size_t)b * LL * CC;
  for (int c = t; c < CC; c += 256) {
    const int h = c >> 6;                    // D = 64
    float a = 0.f;
    #pragma unroll 1
    for (int s = 0; s < SS; ++s)
      a += s_aw[h*SS + s] * xb[(size_t)s_idx[s]*CC + c];
    out_attn[(size_t)bl*CC + c] = a;
  }
}

__global__ void ent_reduce_kernel(const float* __restrict__ ent, float* __restrict__ out)
{
  __shared__ float s[256];
  const int t = threadIdx.x;
  float a = 0.f;
  for (int i = t; i < MM; i += 256) a += ent[i];
  s[t] = a; __syncthreads();
  for (int w = 128; w > 0; w >>= 1) { if (t < w) s[t] += s[t + w]; __syncthreads(); }
  if (t == 0) out[0] = s[0] / (float)(MM * HH);   // mean over (B,L,H) of sum_s aw*log(aw+eps)
}

extern "C" void kernel_launch(void* const* d_in, const int* in_sizes, int n_in,
                              void* d_out, int out_size, void* d_ws, size_t ws_size,
                              hipStream_t stream)
{
  (void)in_sizes; (void)n_in; (void)out_size; (void)ws_size;
  const float* x  = (const float*)d_in[0];
  const float* Ww = (const float*)d_in[1];
  const float* bw = (const float*)d_in[2];
  const float* Wq = (const float*)d_in[3];
  const float* bq = (const float*)d_in[4];
  const float* Wk = (const float*)d_in[5];
  const float* Wo = (const float*)d_in[6];
  const float* W1 = (const float*)d_in[7];
  const float* b1 = (const float*)d_in[8];
  const float* W2 = (const float*)d_in[9];
  const float* b2 = (const float*)d_in[10];
  float* out = (float*)d_out;

  float* ws      = (float*)d_ws;
  float* ws_wave = ws;                            // M*24
  float* ws_q    = ws_wave + (size_t)MM*24;       // M*128
  float* ws_aw   = ws_q    + (size_t)MM*128;      // M*264
  float* ws_ent  = ws_aw   + (size_t)MM*HH*SS;    // M
  float* ws_oa   = ws_ent  + MM;                  // M*512
  float* ws_h1   = ws_oa   + (size_t)MM*CC;       // M*128
  float* ws_se   = ws_h1   + (size_t)MM*128;      // M*512
  int*   ws_idx  = (int*)(ws_se + (size_t)MM*CC); // M*33

  const dim3 blk(256, 1, 1);
  const int mb = MM / 128;                        // 64 M-blocks
  // wave = silu(x @ Ww^T + bw), N=24 (one 64-wide N block, tail-masked)
  gemm_wmma_kernel<1,false,true ><<<dim3(mb, 1), blk, 0, stream>>>(x, nullptr, Ww, bw, ws_wave, MM, 24, CC, 24);
  // queries = silu(x @ Wq^T + bq), N=128 (full tiles)
  gemm_wmma_kernel<1,false,false><<<dim3(mb, 2), blk, 0, stream>>>(x, nullptr, Wq, bq, ws_q,    MM, 128, CC, 128);
  attn_weights_kernel<<<MM, 128, 0, stream>>>(ws_wave, ws_q, Wk, ws_idx, ws_aw, ws_ent);
  gather_kernel<<<MM, 256, 0, stream>>>(x, ws_aw, ws_idx, ws_oa);
  // h1 = silu(out_attn @ W1^T + b1), N=128, K=512
  gemm_wmma_kernel<1,false,false><<<dim3(mb, 2), blk, 0, stream>>>(ws_oa, nullptr, W1, b1, ws_h1, MM, 128, CC, 128);
  // se = sigmoid(h1 @ W2^T + b2), N=512, K=128
  gemm_wmma_kernel<2,false,false><<<dim3(mb, 8), blk, 0, stream>>>(ws_h1, nullptr, W2, b2, ws_se, MM, 512, 128, 512);
  // y = (out_attn .* se) @ Wo^T, N=512, K=512 -> straight to d_out
  gemm_wmma_kernel<0,true ,false><<<dim3(mb, 8), blk, 0, stream>>>(ws_oa, ws_se, Wo, nullptr, out, MM, 512, CC, 512);
  // scalar: -entropy
  ent_reduce_kernel<<<1, 256, 0, stream>>>(ws_ent, out + (size_t)MM*CC);
}